// FeatureSelectionModule_54606214202020
// MI455X (gfx1250) — compile-verified
//
#include <hip/hip_runtime.h>
#include <hip/hip_bf16.h>
#include <stdint.h>

// ---------------- problem constants ----------------
#define B_          16
#define NC_         80
#define A_          33600
#define CH_         84          // 4 box + 80 cls channels in raw_preds
#define TOPK_PRE_   750
#define TOPK_POST_  30
#define NSEL_PAD    768         // 750 padded to multiple of 16/32
#define NT_         48          // 768/16 WMMA tiles per dim
#define NWORDS_     24          // 768/32 bitmask words per row
#define IOU_T       0.4f
#define NROWS_OUT   (B_ * TOPK_POST_)         // 480
#define OFF_REG     ((size_t)NROWS_OUT * 256) // 122880
#define OFF_BOX     (2 * OFF_REG)             // 245760
#define OFF_SC      (OFF_BOX + (size_t)NROWS_OUT * 4) // 247680

typedef __attribute__((ext_vector_type(2))) float v2f;
typedef __attribute__((ext_vector_type(8))) float v8f;

// ---------------- K1: per-anchor max/argmax over 80 classes ----------------
__global__ void k1_maxcls(const float* __restrict__ raw,
                          float* __restrict__ maxs, int* __restrict__ cls) {
    int t = blockIdx.x * blockDim.x + threadIdx.x;
    if (t >= B_ * A_) return;
    int b = t / A_, a = t - b * A_;
    const float* p = raw + (size_t)b * CH_ * A_ + 4 * (size_t)A_ + a;
    float m = -1e30f; int id = 0;
#pragma unroll 4
    for (int c = 0; c < NC_; ++c) {
        float s = p[(size_t)c * A_];     // coalesced across threads
        if (s > m) { m = s; id = c; }    // strict > => first-occurrence argmax
    }
    maxs[t] = m; cls[t] = id;
}

// ---------------- K2: exact top-750 (radix-select + bitonic sort) ----------
__global__ __launch_bounds__(1024)
void k2_select(const float* __restrict__ raw, const float* __restrict__ maxs,
               const int* __restrict__ cls, int* __restrict__ sel_idx,
               float* __restrict__ bx_off) {
    __shared__ unsigned hist[256];
    __shared__ unsigned skey[1024];
    __shared__ int      sidx[1024];
    __shared__ unsigned sh_prefix, sh_k, sh_cnt, sh_maxbits;
    const int b = blockIdx.x, tid = threadIdx.x;
    const float* ms = maxs + (size_t)b * A_;
    if (tid == 0) { sh_prefix = 0u; sh_k = TOPK_PRE_; sh_cnt = 0u; sh_maxbits = 0u; }

    // 4-pass radix select: find exact 750th-largest key (positive floats => bit order)
    for (int shift = 24; shift >= 0; shift -= 8) {
        for (int i = tid; i < 256; i += 1024) hist[i] = 0u;
        __syncthreads();
        unsigned pfx = sh_prefix;
        for (int a = tid; a < A_; a += 1024) {
            unsigned key = __float_as_uint(ms[a]);
            bool ok = (shift == 24) || ((key >> (shift + 8)) == pfx);
            if (ok) atomicAdd(&hist[(key >> shift) & 255u], 1u);
        }
        __syncthreads();
        if (tid == 0) {
            unsigned c = 0, k = sh_k; int d = 255;
            for (; d > 0; --d) { if (c + hist[d] >= k) break; c += hist[d]; }
            sh_k = k - c;
            sh_prefix = (pfx << 8) | (unsigned)d;
        }
        __syncthreads();
    }
    const unsigned T = sh_prefix;

    // collect all keys >= T (750 <= count <= ~750+ties, cap 1024)
    for (int a = tid; a < A_; a += 1024) {
        unsigned key = __float_as_uint(ms[a]);
        if (key >= T) {
            unsigned pos = atomicAdd(&sh_cnt, 1u);
            if (pos < 1024u) { skey[pos] = key; sidx[pos] = a; }
        }
    }
    __syncthreads();
    unsigned cnt = sh_cnt; if (cnt > 1024u) cnt = 1024u;
    if ((unsigned)tid >= cnt) { skey[tid] = 0u; sidx[tid] = 0x7FFFFFFF; }
    __syncthreads();

    // bitonic sort 1024: key desc, tie idx asc (jax top_k tie order)
    for (int k = 2; k <= 1024; k <<= 1) {
        for (int j = k >> 1; j > 0; j >>= 1) {
            int i = tid, l = i ^ j;
            if (l > i) {
                unsigned ka = skey[i], kb = skey[l];
                int ia = sidx[i], ib = sidx[l];
                bool before = (ka > kb) || (ka == kb && ia < ib);
                bool desc = ((i & k) == 0);
                if (desc ? !before : before) {
                    skey[i] = kb; sidx[i] = ib; skey[l] = ka; sidx[l] = ia;
                }
            }
            __syncthreads();
        }
    }

    // gather top-750 boxes, reduce max coordinate (for per-class offset)
    float bx4[4] = {0.f, 0.f, 0.f, 0.f};
    int anchor = 0, cid = 0;
    if (tid < TOPK_PRE_) {
        anchor = sidx[tid];
        const float* rp = raw + (size_t)b * CH_ * A_ + anchor;
        bx4[0] = rp[0];
        bx4[1] = rp[(size_t)A_];
        bx4[2] = rp[2 * (size_t)A_];
        bx4[3] = rp[3 * (size_t)A_];
        cid = cls[(size_t)b * A_ + anchor];
        unsigned mb = __float_as_uint(fmaxf(fmaxf(bx4[0], bx4[1]), fmaxf(bx4[2], bx4[3])));
        atomicMax(&sh_maxbits, mb);   // coords >= 0 => float bits monotonic
    }
    __syncthreads();
    float maxc = __uint_as_float(sh_maxbits);
    if (tid < NSEL_PAD) {
        float off = (tid < TOPK_PRE_) ? (float)cid * (maxc + 1.0f) : 0.0f;
        size_t base = ((size_t)b * NSEL_PAD + tid) * 4;
        bx_off[base + 0] = bx4[0] + off;
        bx_off[base + 1] = bx4[1] + off;
        bx_off[base + 2] = bx4[2] + off;
        bx_off[base + 3] = bx4[3] + off;
        sel_idx[(size_t)b * NSEL_PAD + tid] = (tid < TOPK_PRE_) ? anchor : 0;
    }
}

// ---------------- K3: suppression bit-matrix (WMMA union) + serial NMS -----
__global__ __launch_bounds__(1024)
void k3_nms(const float* __restrict__ bx_off, const int* __restrict__ sel_idx,
            int* __restrict__ final_anchor) {
    extern __shared__ unsigned char smem[];
    unsigned* sup   = (unsigned*)smem;                                  // 768*24 words
    float*    bx    = (float*)(smem + (size_t)NSEL_PAD * NWORDS_ * 4);  // 768*4
    float*    areaS = (float*)((unsigned char*)bx + NSEL_PAD * 16);     // 768
    unsigned* keepS = (unsigned*)((unsigned char*)areaS + NSEL_PAD * 4);// 32
    unsigned* wpref = keepS + 32;                                       // 25

    const int tid = threadIdx.x, b = blockIdx.x;
    const int lane = tid & 31, wave = tid >> 5;

    for (int i = tid; i < NSEL_PAD * NWORDS_; i += 1024) sup[i] = 0u;
    if (tid < NSEL_PAD) {
        size_t base = ((size_t)b * NSEL_PAD + tid) * 4;
        float x1 = bx_off[base], y1 = bx_off[base + 1];
        float x2 = bx_off[base + 2], y2 = bx_off[base + 3];
        bx[tid * 4 + 0] = x1; bx[tid * 4 + 1] = y1;
        bx[tid * 4 + 2] = x2; bx[tid * 4 + 3] = y2;
        areaS[tid] = (x2 - x1) * (y2 - y1);
    }
    __syncthreads();

    // 48x48 tiles of 16x16 pairs; one tile per wave iteration (EXEC all-ones)
    for (int t = wave; t < NT_ * NT_; t += 32) {
        int ti = t / NT_, tj = t - ti * NT_;
        int m0 = ti * 16, n0 = tj * 16;
        int jj = n0 + (lane & 15);
        int half = lane >> 4;
        float jx1 = bx[jj * 4], jy1 = bx[jj * 4 + 1];
        float jx2 = bx[jj * 4 + 2], jy2 = bx[jj * 4 + 3];
        float inter[8];
        v8f c;
#pragma unroll
        for (int r = 0; r < 8; ++r) {
            int ii = m0 + r + half * 8;
            float ix1 = bx[ii * 4], iy1 = bx[ii * 4 + 1];
            float ix2 = bx[ii * 4 + 2], iy2 = bx[ii * 4 + 3];
            float iw = fmaxf(fminf(ix2, jx2) - fmaxf(ix1, jx1), 0.f);
            float ih = fmaxf(fminf(iy2, jy2) - fmaxf(iy1, jy1), 0.f);
            inter[r] = iw * ih;
            c[r] = -inter[r];
        }
        // union[i][j] = area_i + area_j - inter : rank-2 outer product via WMMA
        // A (16x4): col0 = area_i, col1 = 1 ; B (4x16): row0 = 1, row1 = area_j
        v2f av, bv;
        av.x = (lane < 16) ? areaS[m0 + lane] : 0.f;   // K=0 (lanes 16-31: K=2 -> 0)
        av.y = (lane < 16) ? 1.f : 0.f;                // K=1 (K=3 -> 0)
        bv.x = (lane < 16) ? 1.f : 0.f;                // K=0 row of ones
        bv.y = (lane < 16) ? areaS[n0 + lane] : 0.f;   // K=1 row = area_j
        v8f d = __builtin_amdgcn_wmma_f32_16x16x4_f32(
            false, av, false, bv, (short)0, c, false, false);

        unsigned w[8];
#pragma unroll
        for (int r = 0; r < 8; ++r) {
            int ii = m0 + r + half * 8;
            bool s = (inter[r] > IOU_T * d[r]) && (jj > ii) &&
                     (ii < TOPK_PRE_) && (jj < TOPK_PRE_);
            w[r] = (unsigned)(__ballot(s) & 0xFFFFFFFFull);
        }
        if (lane == 0) {
            int wsh = (tj & 1) * 16;
            int wj = tj >> 1;
#pragma unroll
            for (int r = 0; r < 8; ++r) {
                atomicOr(&sup[(m0 + r) * NWORDS_ + wj], (w[r] & 0xFFFFu) << wsh);
                atomicOr(&sup[(m0 + r + 8) * NWORDS_ + wj], (w[r] >> 16) << wsh);
            }
        }
    }
    __syncthreads();

    // serial NMS sweep: single wave, keep mask in registers (24 words / 24 lanes)
    if (wave == 0) {
        unsigned keep = (lane < NWORDS_) ? 0xFFFFFFFFu : 0u;
        for (int i = 0; i < TOPK_PRE_; ++i) {
            unsigned row = (lane < NWORDS_) ? sup[i * NWORDS_ + lane] : 0u;
            unsigned kw = __shfl(keep, i >> 5);
            if ((kw >> (i & 31)) & 1u) keep &= ~row;
        }
        if (lane == NWORDS_ - 1) keep &= ((1u << (TOPK_PRE_ & 31)) - 1u); // mask pad bits
        if (lane < NWORDS_) keepS[lane] = keep;
    }
    __syncthreads();
    if (tid == 0) {
        unsigned c = 0;
        for (int wI = 0; wI < NWORDS_; ++wI) { wpref[wI] = c; c += __popc(keepS[wI]); }
        wpref[NWORDS_] = c;  // n_kept
    }
    __syncthreads();

    // stable partition (kept first, score order) -> first 30
    unsigned nkept = wpref[NWORDS_];
    if (tid < TOPK_PRE_) {
        unsigned word = keepS[tid >> 5];
        unsigned bit = (word >> (tid & 31)) & 1u;
        unsigned rank = wpref[tid >> 5] + __popc(word & ((1u << (tid & 31)) - 1u));
        unsigned pos = bit ? rank : nkept + ((unsigned)tid - rank);
        if (pos < TOPK_POST_)
            final_anchor[b * TOPK_POST_ + pos] = sel_idx[(size_t)b * NSEL_PAD + tid];
    }
}

// ---------------- K4: gather features / boxes / scores --------------------
__global__ void k4_gather(const float* __restrict__ raw, const float* __restrict__ vid,
                          const float* __restrict__ reg, const float* __restrict__ maxs,
                          const int* __restrict__ final_anchor, float* __restrict__ out) {
    int r = blockIdx.x;              // 0..479
    int b = r / TOPK_POST_;
    int anchor = final_anchor[r];
    int t = threadIdx.x;             // 256 threads
    size_t fbase = ((size_t)b * A_ + anchor) * 256;
    out[(size_t)r * 256 + t] = vid[fbase + t];
    out[OFF_REG + (size_t)r * 256 + t] = reg[fbase + t];
    if (t < 4)
        out[OFF_BOX + (size_t)r * 4 + t] =
            raw[(size_t)b * CH_ * A_ + (size_t)t * A_ + anchor];
    if (t == 4)
        out[OFF_SC + r] = maxs[(size_t)b * A_ + anchor];
}

// ---------------- launch ---------------------------------------------------
extern "C" void kernel_launch(void* const* d_in, const int* in_sizes, int n_in,
                              void* d_out, int out_size, void* d_ws, size_t ws_size,
                              hipStream_t stream) {
    const float* raw = (const float*)d_in[0];
    const float* vid = (const float*)d_in[1];
    const float* reg = (const float*)d_in[2];

    char* ws = (char*)d_ws;
    size_t off = 0;
    float* maxs        = (float*)(ws + off); off += (size_t)B_ * A_ * 4;
    int*   cls         = (int*)  (ws + off); off += (size_t)B_ * A_ * 4;
    int*   sel_idx     = (int*)  (ws + off); off += (size_t)B_ * NSEL_PAD * 4;
    float* bx_off      = (float*)(ws + off); off += (size_t)B_ * NSEL_PAD * 16;
    int*   final_anch  = (int*)  (ws + off); off += (size_t)B_ * TOPK_POST_ * 4;

    const size_t k3_lds = (size_t)NSEL_PAD * NWORDS_ * 4   // sup bit matrix
                        + (size_t)NSEL_PAD * 16            // boxes
                        + (size_t)NSEL_PAD * 4             // areas
                        + 32 * 4 + 32 * 4;                 // keep words + prefix
    (void)hipFuncSetAttribute((const void*)k3_nms,
                              hipFuncAttributeMaxDynamicSharedMemorySize,
                              (int)k3_lds);

    int blocks1 = (B_ * A_ + 255) / 256;
    k1_maxcls<<<blocks1, 256, 0, stream>>>(raw, maxs, cls);
    k2_select<<<B_, 1024, 0, stream>>>(raw, maxs, cls, sel_idx, bx_off);
    k3_nms<<<B_, 1024, k3_lds, stream>>>(bx_off, sel_idx, final_anch);
    k4_gather<<<NROWS_OUT, 256, 0, stream>>>(raw, vid, reg, maxs, final_anch,
                                             (float*)d_out);
}